// DDPLL_23106924052733
// MI455X (gfx1250) — compile-verified
//
#include <hip/hip_runtime.h>
#include <stdint.h>

#define NSYMB            40000
#define NMODES           256
#define LEADSYM          20000
#define CHUNK            64
#define NCHUNK           (NSYMB / CHUNK)      // 625
#define MODES_PER_BLOCK  32
#define TILE_ELEMS       (CHUNK * MODES_PER_BLOCK)   // 2048 floats = 8192 B

typedef unsigned int tdm_v4u __attribute__((ext_vector_type(4)));
typedef int          tdm_v8i __attribute__((ext_vector_type(8)));
typedef int          tdm_v4i __attribute__((ext_vector_type(4)));

// Explicit LDS (addrspace 3) pointer type: loads through it lower to ds_load_b32.
typedef __attribute__((address_space(3))) const float lds_cf;

// Build a D# for a 2D [CHUNK x 32] f32 tile out of a [NSYMB x NMODES] row-major
// tensor and kick the Tensor Data Mover (async, tracked by TENSORcnt).
__device__ __forceinline__ void tdm_tile_load(const float* gbase, uint32_t row0,
                                              uint32_t col0, uint32_t lds_off) {
  uint64_t ga = (uint64_t)(uintptr_t)gbase +
                (((uint64_t)row0 * (uint64_t)NMODES + (uint64_t)col0) << 2);
  tdm_v4u g0;
  g0.x = 1u;                                  // count=1 (valid), user mode
  g0.y = lds_off;                             // LDS byte address
  g0.z = (uint32_t)ga;                        // global_addr[31:0]
  g0.w = (uint32_t)(ga >> 32) | (2u << 30);   // global_addr[56:32] | type=2
  tdm_v8i g1;
  g1[0] = (int)(2u << 16);                                // data_size = 4B
  g1[1] = (int)(((uint32_t)NMODES & 0xFFFFu) << 16);      // tensor_dim0[15:0]
  g1[2] = (int)(((uint32_t)NSYMB  & 0xFFFFu) << 16);      // tensor_dim1[15:0]
  g1[3] = (int)((uint32_t)MODES_PER_BLOCK << 16);         // tile_dim0 = 32
  g1[4] = (int)CHUNK;                                     // tile_dim1 = CHUNK
  g1[5] = (int)NMODES;                                    // tensor_dim0_stride
  g1[6] = 0;
  g1[7] = 0;
  tdm_v4i gz4 = {0, 0, 0, 0};                 // 2D tensor: groups 2/3 unused
  tdm_v8i gz8 = {0, 0, 0, 0, 0, 0, 0, 0};     // extra group (6-arg toolchain form)
  __builtin_amdgcn_tensor_load_to_lds(g0, g1, gz4, gz4, gz8, 0);
}

__global__ __launch_bounds__(MODES_PER_BLOCK)
void ddpll_kernel(const float* __restrict__ EiR, const float* __restrict__ EiI,
                  const float* __restrict__ TxR, const float* __restrict__ TxI,
                  const float* __restrict__ eta, float* __restrict__ out) {
  // [buffer][array: EiR,EiI,TxR,TxI][CHUNK*32]  -> 64 KB LDS, TDM-written.
  __shared__ float sm[2][4][TILE_ELEMS];

  const int lane      = threadIdx.x;                       // 0..31, wave32
  const int mode      = blockIdx.x * MODES_PER_BLOCK + lane;
  const uint32_t col0 = blockIdx.x * MODES_PER_BLOCK;
  const uint32_t smbase = (uint32_t)(uintptr_t)&sm[0][0][0]; // LDS byte offset
  const uint32_t abytes = TILE_ELEMS * 4u;

  // Launder the LDS offset: compiler loses provenance, so it cannot fold the
  // reads of a "never-written" buffer (TDM writes it behind the compiler's
  // back), and an addrspace(3) pointer built from it yields native ds_load_b32.
  uint32_t smoff = smbase;
  asm volatile("" : "+s"(smoff));
  lds_cf* lp = (lds_cf*)smoff;   // AS3 pointers are 32-bit on amdgcn

  const float Kv = tanhf(eta[0]);

  const float PERIOD   = 1.57079632679489662f;  // pi/2 (unwrap period)
  const float HALFP    = 0.78539816339744831f;  // pi/4
  const float INV_SQ10 = 0.316227766016837933f; // 1/sqrt(10)
  const float HSQ10    = 1.58113883008418966f;  // sqrt(10)/2
  const size_t THETA_OFF = (size_t)NSYMB * NMODES * 2;

  // Prime the double buffer: chunk 0 -> buf 0, chunk 1 -> buf 1 (8 TDM ops in flight).
  {
    uint32_t b0 = smbase;
    tdm_tile_load(EiR, 0, col0, b0 + 0u * abytes);
    tdm_tile_load(EiI, 0, col0, b0 + 1u * abytes);
    tdm_tile_load(TxR, 0, col0, b0 + 2u * abytes);
    tdm_tile_load(TxI, 0, col0, b0 + 3u * abytes);
    uint32_t b1 = smbase + 4u * abytes;
    tdm_tile_load(EiR, CHUNK, col0, b1 + 0u * abytes);
    tdm_tile_load(EiI, CHUNK, col0, b1 + 1u * abytes);
    tdm_tile_load(TxR, CHUNK, col0, b1 + 2u * abytes);
    tdm_tile_load(TxI, CHUNK, col0, b1 + 3u * abytes);
  }

  // Per-mode recurrence state (u[1] of the reference is write-only -> dropped).
  float phi = 0.f, u0 = 0.f, gp = 0.f;
  float thp = 0.f, cum = 0.f;   // unwrap state: previous raw theta, cumulative correction

  for (int c = 0; c < NCHUNK; ++c) {
    // Retire this chunk's 4 TDM loads (in-order per wave): after the wait,
    // at most the *next* chunk's 4 remain outstanding.
    if (c + 1 < NCHUNK) __builtin_amdgcn_s_wait_tensorcnt((unsigned short)4);
    else                __builtin_amdgcn_s_wait_tensorcnt((unsigned short)0);
    asm volatile("" ::: "memory");

    const int b = c & 1;
    const uint32_t bufe = (uint32_t)b * 4u * TILE_ELEMS;   // element offset of buffer

#pragma unroll 4
    for (int kk = 0; kk < CHUNK; ++kk) {
      const int k  = c * CHUNK + kk;
      const uint32_t li = bufe + (uint32_t)(kk * MODES_PER_BLOCK + lane);

      float yre = lp[li + 0u * TILE_ELEMS];   // ds_load_b32, offset imm 0
      float yim = lp[li + 1u * TILE_ELEMS];   // ds_load_b32, offset imm 8192

      const float th = phi;   // emitted theta is the PRE-update phase

      // ---- streaming jnp.unwrap(period=pi/2) ----
      float dd = th - thp;
      float r  = fmodf(dd + HALFP, PERIOD);
      r = (r < 0.f) ? r + PERIOD : r;
      r -= HALFP;
      if (r == -HALFP && dd > 0.f) r = HALFP;
      float corr = r - dd;
      if (fabsf(dd) < HALFP) corr = 0.f;
      cum += corr;
      thp = th;
      const float thu = th + cum;

      // ---- outputs: Eo = Ei * exp(i*theta_unwrapped), theta ----
      float su, cu;
      __sincosf(thu, &su, &cu);
      float2 eo;
      eo.x = yre * cu - yim * su;
      eo.y = yre * su + yim * cu;
      *(float2*)(out + (((size_t)k * NMODES + mode) << 1)) = eo;
      out[THETA_OFF + (size_t)k * NMODES + mode] = thu;

      // ---- gradient gk = 2*Im(z * conj(target)), z = y*exp(i*phi) ----
      float sp, cp;
      __sincosf(th, &sp, &cp);
      const float zre = yre * cp - yim * sp;
      const float zim = yre * sp + yim * cp;

      float tre, tim;
      if (k >= LEADSYM) {  // decision-directed: nearest 16-QAM point (separable)
        float ir = fminf(fmaxf(floorf(zre * HSQ10 + 2.0f), 0.f), 3.f);
        float ii = fminf(fmaxf(floorf(zim * HSQ10 + 2.0f), 0.f), 3.f);
        tre = (2.f * ir - 3.f) * INV_SQ10;
        tim = (2.f * ii - 3.f) * INV_SQ10;
      } else {             // pilot-aided
        tre = lp[li + 2u * TILE_ELEMS];       // ds_load_b32, offset imm 16384
        tim = lp[li + 3u * TILE_ELEMS];       // ds_load_b32, offset imm 24576
      }
      const float gk = 2.f * (zim * tre - zre * tim);

      // momentum filter: u0 <- u0 - g_prev + gk ; phi <- phi - Kv*u0
      const float u0n = u0 - gp + gk;
      gp  = gk;
      u0  = u0n;
      phi = phi - Kv * u0n;
    }

    // Refill the buffer we just drained with chunk c+2.
    if (c + 2 < NCHUNK) {
      asm volatile("s_wait_dscnt 0" ::: "memory");  // LDS reads done before TDM overwrites
      const uint32_t row0 = (uint32_t)(c + 2) * CHUNK;
      const uint32_t bb   = smbase + (uint32_t)b * 4u * abytes;
      tdm_tile_load(EiR, row0, col0, bb + 0u * abytes);
      tdm_tile_load(EiI, row0, col0, bb + 1u * abytes);
      tdm_tile_load(TxR, row0, col0, bb + 2u * abytes);
      tdm_tile_load(TxI, row0, col0, bb + 3u * abytes);
    }
  }
}

extern "C" void kernel_launch(void* const* d_in, const int* in_sizes, int n_in,
                              void* d_out, int out_size, void* d_ws, size_t ws_size,
                              hipStream_t stream) {
  (void)in_sizes; (void)n_in; (void)out_size; (void)d_ws; (void)ws_size;
  const float* EiR = (const float*)d_in[0];
  const float* EiI = (const float*)d_in[1];
  const float* TxR = (const float*)d_in[2];
  const float* TxI = (const float*)d_in[3];
  const float* eta = (const float*)d_in[4];
  float* out = (float*)d_out;

  dim3 grid(NMODES / MODES_PER_BLOCK);   // 8 blocks -> 8 independent waves/SIMDs
  dim3 block(MODES_PER_BLOCK);           // one wave32 per block owns its TDM pipeline
  hipLaunchKernelGGL(ddpll_kernel, grid, block, 0, stream, EiR, EiI, TxR, TxI, eta, out);
}